// ClusterGCNConvNet_20968030339123
// MI455X (gfx1250) — compile-verified
//
#include <hip/hip_runtime.h>

// ---------------------------------------------------------------------------
// ClusterGCN 3-layer forward for MI455X (gfx1250, wave32, WMMA)
//   per layer: agg = deg_inv * scatter_add(x[src] -> dst)
//              h   = relu(agg @ W_out + b + x @ W_root)      (last layer: no relu)
//   final: log_softmax over 10 classes
// GEMM phase uses V_WMMA_F32_16X16X4_F32 (full fp32, K tiled by 4).
// Weight LDS layout is K-pair interleaved + bank-staggered so every B fragment
// is a single conflict-free ds_load_b64. Epilogue has a wave-uniform fast path
// (only the last block's two out-of-range waves take the predicated store path).
// ---------------------------------------------------------------------------

#define N_NODES 100000
#define N_EDGES 1600000

typedef __attribute__((ext_vector_type(2))) float v2f;
typedef __attribute__((ext_vector_type(8))) float v8f;

__device__ __forceinline__ v8f wmma4(v2f a, v2f b, v8f c) {
  // D = A(16x4) * B(4x16) + C(16x16), fp32
  return __builtin_amdgcn_wmma_f32_16x16x4_f32(
      /*neg_a=*/false, a, /*neg_b=*/false, b,
      /*c_mod=*/(short)0, c, /*reuse_a=*/false, /*reuse_b=*/false);
}

// -------------------------------- utility ----------------------------------

__global__ void zero_kernel(float* __restrict__ p, int n) {
  int i = blockIdx.x * blockDim.x + threadIdx.x;
  if (i < n) p[i] = 0.0f;
}

__global__ void degree_kernel(const int* __restrict__ dst, float* __restrict__ deg) {
  int e = blockIdx.x * blockDim.x + threadIdx.x;
  if (e < N_EDGES) atomicAdd(&deg[dst[e]], 1.0f);
}

__global__ void inv_kernel(float* __restrict__ deg) {
  int i = blockIdx.x * blockDim.x + threadIdx.x;
  if (i < N_NODES) deg[i] = 1.0f / fmaxf(deg[i], 1.0f);
}

// agg[dst] += x[src]; 16 threads per edge, float4 per thread, f32 atomics.
__global__ void scatter_kernel(const float* __restrict__ xin,
                               const int* __restrict__ src,
                               const int* __restrict__ dst,
                               float* __restrict__ agg) {
  int tid = blockIdx.x * blockDim.x + threadIdx.x;
  if (tid >= N_EDGES * 16) return;
  int e = tid >> 4;
  int c = (tid & 15) << 2;
  int s = src[e];
  int d = dst[e];
  const float4 v = *reinterpret_cast<const float4*>(xin + (size_t)s * 64 + c);
  float* p = agg + (size_t)d * 64 + c;
  atomicAdd(p + 0, v.x);
  atomicAdd(p + 1, v.y);
  atomicAdd(p + 2, v.z);
  atomicAdd(p + 3, v.w);
}

// ------------------------------ fused GEMM ---------------------------------
// out[row, col] = act( (deginv[row]*agg[row,:]) @ Wout + bias + xin[row,:] @ Wroot )
// K = 64 fixed. CT = 16-wide column tiles (4 -> 64 cols, 1 -> 10 padded to 16).
// Block = 128 threads = 4 waves; each wave computes a 16-row slab.
template <int CT, int OUTC, bool RELU>
__global__ __launch_bounds__(128) void gcn_gemm_kernel(
    const float* __restrict__ agg, const float* __restrict__ xin,
    const float* __restrict__ deginv,
    const float* __restrict__ Wout,   // [64 x OUTC], row-major (in, out)
    const float* __restrict__ bias,   // [OUTC]
    const float* __restrict__ Wroot,  // [64 x OUTC]
    float* __restrict__ out) {
  constexpr int PADC = CT * 16;
  constexpr int AP = 68;  // A-tile row pitch: banks 4*l16 + 2*half -> conflict-free b64
  // Weight group pitch (floats) for one K-pair group of PADC cols, pair-interleaved.
  // Require P2 % 64 == 32 so half=1 lanes land on banks 32..63 (conflict-free b64).
  constexpr int P2 = (2 * PADC) + (((2 * PADC) % 64 == 32) ? 0 : 32);

  __shared__ float sWo[32 * P2];  // [k>>1][col*2 + (k&1)]
  __shared__ float sWr[32 * P2];
  __shared__ float sBias[PADC];
  __shared__ float sA[4 * 16 * AP];
  __shared__ float sX[4 * 16 * AP];

  const int tid = threadIdx.x;
  const int wave = tid >> 5;
  const int lane = tid & 31;
  const int l16 = lane & 15;
  const int laneHi = lane >> 4;  // 0: K/M low half, 1: high half

  // ---- stage weights (zero-padded cols, K-pair interleaved) into LDS ----
  for (int idx = tid; idx < 64 * PADC; idx += 128) {
    int k = idx / PADC;
    int n = idx % PADC;
    int di = (k >> 1) * P2 + n * 2 + (k & 1);
    float wo = (n < OUTC) ? Wout[k * OUTC + n] : 0.0f;
    float wr = (n < OUTC) ? Wroot[k * OUTC + n] : 0.0f;
    sWo[di] = wo;
    sWr[di] = wr;
  }
  if (tid < PADC) sBias[tid] = (tid < OUTC) ? bias[tid] : 0.0f;

  // ---- each wave stages its 16x64 tiles of (scaled agg) and xin ----
  const int rowBase = blockIdx.x * 64 + wave * 16;
  float* tA = sA + wave * 16 * AP;
  float* tX = sX + wave * 16 * AP;
#pragma unroll
  for (int i = 0; i < 8; ++i) {
    int f = i * 128 + lane * 4;  // flat float index in 16x64 tile
    int r = f >> 6;
    int c = f & 63;
    int gr = rowBase + r;
    if (gr >= N_NODES) gr = N_NODES - 1;  // clamp (stores predicated later)
    float sc = deginv[gr];
    const float4 va = *reinterpret_cast<const float4*>(agg + (size_t)gr * 64 + c);
    const float4 vx = *reinterpret_cast<const float4*>(xin + (size_t)gr * 64 + c);
    float* pa = tA + r * AP + c;
    pa[0] = va.x * sc; pa[1] = va.y * sc; pa[2] = va.z * sc; pa[3] = va.w * sc;
    float* px = tX + r * AP + c;
    px[0] = vx.x; px[1] = vx.y; px[2] = vx.z; px[3] = vx.w;
  }
  __syncthreads();

  // ---- WMMA main loop: 16 K-steps of 4, two GEMMs accumulated together ----
  v8f acc[CT] = {};
  for (int ks = 0; ks < 16; ++ks) {
    const int kk = ks * 4 + laneHi * 2;  // A 16x4 layout: lanes 16-31 hold K+2
    const int g = ks * 2 + laneHi;       // weight K-pair group index
    // A fragments: contiguous K pair -> single ds_load_b64
    v2f a = *reinterpret_cast<const v2f*>(tA + l16 * AP + kk);
    v2f ax = *reinterpret_cast<const v2f*>(tX + l16 * AP + kk);
#pragma unroll
    for (int t = 0; t < CT; ++t) {
      const int col = t * 16 + l16;
      // B fragments: pair-interleaved layout -> single conflict-free ds_load_b64
      v2f bo = *reinterpret_cast<const v2f*>(sWo + g * P2 + col * 2);
      acc[t] = wmma4(a, bo, acc[t]);
      v2f br = *reinterpret_cast<const v2f*>(sWr + g * P2 + col * 2);
      acc[t] = wmma4(ax, br, acc[t]);
    }
  }

  // ---- epilogue: bias (+ReLU) ----
  // C/D layout: lanes 0-15 -> M=v, N=lane; lanes 16-31 -> M=v+8, N=lane-16
  float bcol[CT];
  bool colOk[CT];
#pragma unroll
  for (int t = 0; t < CT; ++t) {
    const int col = t * 16 + l16;
    colOk[t] = (col < OUTC);
    bcol[t] = sBias[col];
#pragma unroll
    for (int v = 0; v < 8; ++v) {
      float o = acc[t][v] + bcol[t];
      if (RELU) o = fmaxf(o, 0.0f);
      acc[t][v] = o;
    }
  }

  // Wave-uniform fast path: all 16 rows of this wave are in range for every
  // wave-tile except the last two waves of the grid, which have no rows in range.
  if (rowBase + 16 <= N_NODES) {
#pragma unroll
    for (int t = 0; t < CT; ++t) {
      if (CT == 1 ? colOk[t] : true) {  // OUTC==64 tiles have all cols valid
        if (colOk[t]) {
          const int col = t * 16 + l16;
          float* op = out + (size_t)(rowBase + laneHi * 8) * OUTC + col;
#pragma unroll
          for (int v = 0; v < 8; ++v) op[(size_t)v * OUTC] = acc[t][v];
        }
      }
    }
  } else {
#pragma unroll
    for (int t = 0; t < CT; ++t) {
      const int col = t * 16 + l16;
#pragma unroll
      for (int v = 0; v < 8; ++v) {
        const int row = rowBase + v + laneHi * 8;
        if (row < N_NODES && colOk[t]) out[(size_t)row * OUTC + col] = acc[t][v];
      }
    }
  }
}

// ------------------------------ log_softmax --------------------------------
__global__ void logsoftmax_kernel(const float* __restrict__ logits,
                                  float* __restrict__ out) {
  int i = blockIdx.x * blockDim.x + threadIdx.x;
  if (i >= N_NODES) return;
  const float* r = logits + (size_t)i * 10;
  float v[10];
  float m = r[0];
#pragma unroll
  for (int j = 0; j < 10; ++j) {
    v[j] = r[j];
    m = fmaxf(m, v[j]);
  }
  float s = 0.0f;
#pragma unroll
  for (int j = 0; j < 10; ++j) {
    v[j] -= m;
    s += expf(v[j]);
  }
  float ls = logf(s);
  float* o = out + (size_t)i * 10;
#pragma unroll
  for (int j = 0; j < 10; ++j) o[j] = v[j] - ls;
}

// ------------------------------- launcher ----------------------------------
extern "C" void kernel_launch(void* const* d_in, const int* in_sizes, int n_in,
                              void* d_out, int out_size, void* d_ws, size_t ws_size,
                              hipStream_t stream) {
  const float* x   = (const float*)d_in[0];
  const int*   ei  = (const int*)d_in[1];
  const float* Wo0 = (const float*)d_in[2];
  const float* b0  = (const float*)d_in[3];
  const float* Wr0 = (const float*)d_in[4];
  const float* Wo1 = (const float*)d_in[5];
  const float* b1  = (const float*)d_in[6];
  const float* Wr1 = (const float*)d_in[7];
  const float* Wo2 = (const float*)d_in[8];
  const float* b2  = (const float*)d_in[9];
  const float* Wr2 = (const float*)d_in[10];

  const int* src = ei;            // edge_index[0]
  const int* dst = ei + N_EDGES;  // edge_index[1]

  // workspace layout (floats): deg_inv | agg | h0 | h1 | logits
  float* ws     = (float*)d_ws;
  float* deg    = ws;
  float* agg    = deg + N_NODES;
  float* h0     = agg + (size_t)N_NODES * 64;
  float* h1     = h0 + (size_t)N_NODES * 64;
  float* logits = h1 + (size_t)N_NODES * 64;
  float* outf   = (float*)d_out;

  auto cdiv = [](long a, long b) { return (int)((a + b - 1) / b); };
  const int NF = N_NODES * 64;  // 6.4M
  const int ST = N_EDGES * 16;  // 25.6M scatter threads
  const int gemmGrid = cdiv(N_NODES, 64);

  // degrees (shared by all layers)
  zero_kernel<<<cdiv(N_NODES, 256), 256, 0, stream>>>(deg, N_NODES);
  degree_kernel<<<cdiv(N_EDGES, 256), 256, 0, stream>>>(dst, deg);
  inv_kernel<<<cdiv(N_NODES, 256), 256, 0, stream>>>(deg);

  // ---- layer 0: x -> h0 ----
  zero_kernel<<<cdiv(NF, 256), 256, 0, stream>>>(agg, NF);
  scatter_kernel<<<cdiv(ST, 256), 256, 0, stream>>>(x, src, dst, agg);
  gcn_gemm_kernel<4, 64, true><<<gemmGrid, 128, 0, stream>>>(agg, x, deg, Wo0, b0, Wr0, h0);

  // ---- layer 1: h0 -> h1 ----
  zero_kernel<<<cdiv(NF, 256), 256, 0, stream>>>(agg, NF);
  scatter_kernel<<<cdiv(ST, 256), 256, 0, stream>>>(h0, src, dst, agg);
  gcn_gemm_kernel<4, 64, true><<<gemmGrid, 128, 0, stream>>>(agg, h0, deg, Wo1, b1, Wr1, h1);

  // ---- layer 2: h1 -> logits (10 cols, no relu) ----
  zero_kernel<<<cdiv(NF, 256), 256, 0, stream>>>(agg, NF);
  scatter_kernel<<<cdiv(ST, 256), 256, 0, stream>>>(h1, src, dst, agg);
  gcn_gemm_kernel<1, 10, false><<<gemmGrid, 128, 0, stream>>>(agg, h1, deg, Wo2, b2, Wr2, logits);

  // ---- log_softmax -> d_out ----
  logsoftmax_kernel<<<cdiv(N_NODES, 256), 256, 0, stream>>>(logits, outf);
}